// WidePrecision_28226525070228
// MI455X (gfx1250) — compile-verified
//
#include <hip/hip_runtime.h>
#include <hip/hip_bf16.h>

// out = clip((TP+1)/(S+1), 0, 1)  with  TP = sum(inputs * dilate3x3(targets)),
// S = sum(inputs).  (fp+tp == S exactly, so fp is never needed.)

typedef __attribute__((ext_vector_type(2))) float v2f;
typedef __attribute__((ext_vector_type(4))) float v4f;
typedef __attribute__((ext_vector_type(8))) float v8f;

#define HH 4096
#define WW 8192
#define ROWS_PER_BLOCK 32
#define GRID_X 8            // 8 * 256 threads * 4 cols = 8192 columns
#define GRID_Y (HH / ROWS_PER_BLOCK)
#define NEG_INF (-__builtin_inff())

// Horizontal max-of-3 over this thread's 4 columns of row r.
// Column halo comes from neighbor lanes via wave32 shuffles; wave-edge lanes
// (and image borders -> -inf identity) fall back to scalar loads / -inf.
__device__ __forceinline__ v4f load_hmax(const float* __restrict__ tg,
                                         int r, int col0, int lane) {
  const v4f t = *(const v4f*)(tg + (size_t)r * WW + col0);
  float left  = __shfl_up(t.w, 1);
  float right = __shfl_down(t.x, 1);
  if (lane == 0)
    left = (col0 == 0) ? NEG_INF : tg[(size_t)r * WW + col0 - 1];
  if (lane == 31)
    right = (col0 + 4 >= WW) ? NEG_INF : tg[(size_t)r * WW + col0 + 4];
  v4f h;
  h.x = fmaxf(fmaxf(left, t.x), t.y);
  h.y = fmaxf(fmaxf(t.x, t.y), t.z);
  h.z = fmaxf(fmaxf(t.y, t.z), t.w);
  h.w = fmaxf(fmaxf(t.z, t.w), right);
  return h;
}

// Wave-level sum of one f32 per lane via V_WMMA_F32_16X16X4_F32.
// A = all-ones 16x4 (layout-independent), B holds the 32 partials in one of
// its two VGPRs (the other is zero).  Every D element equals a column-sum of
// B; D vgpr0 over lanes 0..15 enumerates all 16 column sums (documented C/D
// layout), so xor-reducing d[0] over masks 1,2,4,8 yields sum(B) exactly.
__device__ __forceinline__ float wave_sum_wmma(float x) {
  v2f a;  a[0] = 1.0f; a[1] = 1.0f;
  v2f b;  b[0] = x;    b[1] = 0.0f;
  v8f c = {};
  v8f d = __builtin_amdgcn_wmma_f32_16x16x4_f32(false, a, false, b,
                                                (short)0, c, false, false);
  float r = d[0];
  r += __shfl_xor(r, 1);
  r += __shfl_xor(r, 2);
  r += __shfl_xor(r, 4);
  r += __shfl_xor(r, 8);
  return r;   // identical in all 32 lanes (both 16-lane halves duplicate D)
}

__global__ __launch_bounds__(256)
void wide_precision_partial(const float* __restrict__ in,
                            const float* __restrict__ tg,
                            float2* __restrict__ partial) {
  const int lane = threadIdx.x & 31;
  const int col0 = blockIdx.x * 1024 + threadIdx.x * 4;
  const int r0   = blockIdx.y * ROWS_PER_BLOCK;

  const v4f ninf = {NEG_INF, NEG_INF, NEG_INF, NEG_INF};

  // Rolling 3-row window of horizontal maxes (vertical part of the dilation).
  v4f hm_m1 = (r0 > 0) ? load_hmax(tg, r0 - 1, col0, lane) : ninf;
  v4f hm_0  = load_hmax(tg, r0, col0, lane);

  float tp = 0.0f, s = 0.0f;

#pragma unroll 4
  for (int i = 0; i < ROWS_PER_BLOCK; ++i) {
    const int r = r0 + i;

    // Prefetch upcoming rows (global_prefetch_b8).
    const int prT = (r + 2 < HH) ? r + 2 : HH - 1;
    const int prI = (r + 1 < HH) ? r + 1 : HH - 1;
    __builtin_prefetch(tg + (size_t)prT * WW + col0, 0, 3);
    __builtin_prefetch(in + (size_t)prI * WW + col0, 0, 0);

    const v4f hm_p1 = (r + 1 < HH) ? load_hmax(tg, r + 1, col0, lane) : ninf;

    v4f td;
    td.x = fmaxf(fmaxf(hm_m1.x, hm_0.x), hm_p1.x);
    td.y = fmaxf(fmaxf(hm_m1.y, hm_0.y), hm_p1.y);
    td.z = fmaxf(fmaxf(hm_m1.z, hm_0.z), hm_p1.z);
    td.w = fmaxf(fmaxf(hm_m1.w, hm_0.w), hm_p1.w);

    // inputs are read exactly once -> non-temporal load.
    const v4f iv =
        __builtin_nontemporal_load((const v4f*)(in + (size_t)r * WW + col0));

    tp = fmaf(iv.x, td.x, tp);
    tp = fmaf(iv.y, td.y, tp);
    tp = fmaf(iv.z, td.z, tp);
    tp = fmaf(iv.w, td.w, tp);
    s += (iv.x + iv.y) + (iv.z + iv.w);

    hm_m1 = hm_0;
    hm_0  = hm_p1;
  }

  // Wave reduction through the matrix pipe (EXEC is all ones here: the grid
  // tiles the 4096x8192 domain exactly, no early exits).
  const float wtp = wave_sum_wmma(tp);
  const float wss = wave_sum_wmma(s);

  __shared__ float2 smem[8];
  const int wave = threadIdx.x >> 5;
  if (lane == 0) smem[wave] = make_float2(wtp, wss);
  __syncthreads();
  if (threadIdx.x == 0) {
    float btp = 0.0f, bss = 0.0f;
#pragma unroll
    for (int w = 0; w < 8; ++w) { btp += smem[w].x; bss += smem[w].y; }
    partial[blockIdx.y * gridDim.x + blockIdx.x] = make_float2(btp, bss);
  }
}

__global__ __launch_bounds__(256)
void wide_precision_final(const float2* __restrict__ partial,
                          float* __restrict__ out, int nblocks) {
  float tp = 0.0f, s = 0.0f;
  for (int i = threadIdx.x; i < nblocks; i += 256) {
    const float2 p = partial[i];
    tp += p.x;
    s  += p.y;
  }
#pragma unroll
  for (int m = 1; m <= 16; m <<= 1) {
    tp += __shfl_xor(tp, m);
    s  += __shfl_xor(s, m);
  }
  __shared__ float2 sm[8];
  const int wave = threadIdx.x >> 5;
  const int lane = threadIdx.x & 31;
  if (lane == 0) sm[wave] = make_float2(tp, s);
  __syncthreads();
  if (threadIdx.x == 0) {
    float T = 0.0f, S = 0.0f;
#pragma unroll
    for (int w = 0; w < 8; ++w) { T += sm[w].x; S += sm[w].y; }
    float prec = (T + 1.0f) / (S + 1.0f);
    prec = fminf(fmaxf(prec, 0.0f), 1.0f);
    out[0] = prec;
  }
}

extern "C" void kernel_launch(void* const* d_in, const int* in_sizes, int n_in,
                              void* d_out, int out_size, void* d_ws, size_t ws_size,
                              hipStream_t stream) {
  const float* inputs  = (const float*)d_in[0];
  const float* targets = (const float*)d_in[1];
  float* out = (float*)d_out;
  float2* partial = (float2*)d_ws;   // GRID_X*GRID_Y float2 = 8 KB

  dim3 grid(GRID_X, GRID_Y);
  wide_precision_partial<<<grid, 256, 0, stream>>>(inputs, targets, partial);
  wide_precision_final<<<1, 256, 0, stream>>>(partial, out, GRID_X * GRID_Y);
}